// BridgedSTGNN_3496103379584
// MI455X (gfx1250) — compile-verified
//
#include <hip/hip_runtime.h>

typedef __attribute__((ext_vector_type(2))) float v2f;
typedef __attribute__((ext_vector_type(8))) float v8f;

#define N_NODES 50000
#define N_EDGES 800000
#define D       128
#define OUTD    12
#define N_TGT   8192

// ---------------- degree / normalization (graph-constant, computed once) ----

__global__ void k_zero(float* __restrict__ p, int n) {
  int i = blockIdx.x * blockDim.x + threadIdx.x;
  if (i < n) p[i] = 0.0f;
}

__global__ void k_deg(const int* __restrict__ dst, float* __restrict__ cnt) {
  int e = blockIdx.x * blockDim.x + threadIdx.x;
  if (e < N_EDGES) atomicAdd(&cnt[dst[e]], 1.0f);
}

__global__ void k_dinv(const float* __restrict__ cnt, float* __restrict__ dinv) {
  int i = blockIdx.x * blockDim.x + threadIdx.x;
  if (i < N_NODES) dinv[i] = rsqrtf(cnt[i] + 1.0f);  // +1 for self-loop
}

// ---------------- dense GEMM: H[N x 128] = act(X)[N x 128] @ W[128 x 128] ----
// One block = 16 rows. 8 waves; wave w computes the 16x16 tile for columns
// [16w, 16w+16) via 32 chained V_WMMA_F32_16X16X4_F32 (K = 128 = 32 * 4).
// X tile staged in LDS with optional fused ReLU; padded to 132 floats/row so
// the strided A-fragment reads are bank-conflict free.

__global__ __launch_bounds__(256) void k_gemm(const float* __restrict__ X,
                                              const float* __restrict__ W,
                                              float* __restrict__ H,
                                              int apply_relu) {
  __shared__ float sX[16][132];
  const int row0 = blockIdx.x * 16;

  for (int idx = threadIdx.x; idx < 16 * 128; idx += 256) {
    const int r = idx >> 7, c = idx & 127;
    float v = X[(size_t)(row0 + r) * D + c];
    if (apply_relu) v = fmaxf(v, 0.0f);
    sX[r][c] = v;
  }
  __syncthreads();

  const int wave    = threadIdx.x >> 5;   // 0..7
  const int lane    = threadIdx.x & 31;
  const int half    = lane >> 4;          // 0: K=0,1  1: K=2,3 (fp32 A layout)
  const int l       = lane & 15;          // row (A) / col (B) index
  const int colBase = wave * 16;

  v8f acc = {};
  #pragma unroll
  for (int k = 0; k < 32; ++k) {
    const int kb = 4 * k + 2 * half;
    v2f a, b;
    // A fragment 16x4: lanes 0-15 -> K=kb,kb+1 of row l; lanes 16-31 -> K=kb+2,kb+3
    a.x = sX[l][kb];
    a.y = sX[l][kb + 1];
    // B fragment 4x16: row striped across lanes within a VGPR (mirrors A halves)
    b.x = W[(size_t)kb       * D + colBase + l];
    b.y = W[(size_t)(kb + 1) * D + colBase + l];
    acc = __builtin_amdgcn_wmma_f32_16x16x4_f32(
        /*neg_a=*/false, a, /*neg_b=*/false, b,
        /*c_mod=*/(short)0, acc, /*reuse_a=*/false, /*reuse_b=*/false);
  }

  // D layout: VGPR r -> lanes 0-15: M=r, N=l ; lanes 16-31: M=r+8, N=l
  #pragma unroll
  for (int r = 0; r < 8; ++r) {
    H[(size_t)(row0 + r + 8 * half) * D + colBase + l] = acc[r];
  }
}

// ---------------- aggregation ------------------------------------------------
// A[i] = H[i]*dinv[i]^2 + bias   (self-loop + bias, full overwrite)

__global__ void k_init(const float* __restrict__ H, const float* __restrict__ dinv,
                       const float* __restrict__ bias, float* __restrict__ A) {
  const int i    = blockIdx.x * blockDim.x + threadIdx.x;  // over N*D exactly
  const int node = i >> 7;
  const int f    = i & 127;
  const float di = dinv[node];
  A[i] = H[i] * di * di + bias[f];
}

// One wave32 per edge: float4 gather of H[src] (L2-resident: 25.6 MB << 192 MB
// L2), scaled scatter-add into A[dst] with global_atomic_add_f32.

__global__ void k_edge(const float* __restrict__ H, const int* __restrict__ src,
                       const int* __restrict__ dst, const float* __restrict__ dinv,
                       float* __restrict__ A) {
  const int g    = blockIdx.x * blockDim.x + threadIdx.x;
  const int e    = g >> 5;
  const int lane = g & 31;
  if (e >= N_EDGES) return;
  const int s = src[e];
  const int d = dst[e];
  const float norm = dinv[s] * dinv[d];
  const float4 hv = ((const float4*)(H + (size_t)s * D))[lane];
  float* ap = A + (size_t)d * D + lane * 4;
  atomicAdd(ap + 0, hv.x * norm);
  atomicAdd(ap + 1, hv.y * norm);
  atomicAdd(ap + 2, hv.z * norm);
  atomicAdd(ap + 3, hv.w * norm);
}

// ---------------- regressor head: pred = relu(A)[tgt] @ Wr + br --------------
// One wave32 per target node; wave shuffle reduction. 12.6 MFLOP total.

__global__ void k_final(const float* __restrict__ A, const int* __restrict__ tgt,
                        const float* __restrict__ Wr, const float* __restrict__ br,
                        float* __restrict__ out) {
  const int g    = blockIdx.x * blockDim.x + threadIdx.x;
  const int t    = g >> 5;
  const int lane = g & 31;
  if (t >= N_TGT) return;
  const int node = tgt[t];
  float h[4];
  #pragma unroll
  for (int j = 0; j < 4; ++j)
    h[j] = fmaxf(A[(size_t)node * D + lane + 32 * j], 0.0f);
  for (int o = 0; o < OUTD; ++o) {
    float p = 0.0f;
    #pragma unroll
    for (int j = 0; j < 4; ++j)
      p += h[j] * Wr[(size_t)(lane + 32 * j) * OUTD + o];
    for (int m = 16; m >= 1; m >>= 1) p += __shfl_xor(p, m, 32);
    if (lane == 0) out[(size_t)t * OUTD + o] = p + br[o];
  }
}

// ---------------- driver -----------------------------------------------------

extern "C" void kernel_launch(void* const* d_in, const int* in_sizes, int n_in,
                              void* d_out, int out_size, void* d_ws, size_t ws_size,
                              hipStream_t stream) {
  const float* x   = (const float*)d_in[0];
  const int*   ei  = (const int*)  d_in[1];
  const int*   tgt = (const int*)  d_in[2];
  const float* W1  = (const float*)d_in[3];
  const float* b1  = (const float*)d_in[4];
  const float* W2  = (const float*)d_in[5];
  const float* b2  = (const float*)d_in[6];
  const float* W3  = (const float*)d_in[7];
  const float* b3  = (const float*)d_in[8];
  const float* Wr  = (const float*)d_in[9];
  const float* br  = (const float*)d_in[10];
  float* out = (float*)d_out;

  const int* src = ei;             // edge_index row 0
  const int* dst = ei + N_EDGES;   // edge_index row 1

  // workspace layout (floats): H[N*D] | A[N*D] | cnt[N] | dinv[N]  (~51.6 MB)
  float* ws   = (float*)d_ws;
  float* H    = ws;
  float* A    = H + (size_t)N_NODES * D;
  float* cnt  = A + (size_t)N_NODES * D;
  float* dinv = cnt + N_NODES;

  // normalization (once; graph is layer-invariant)
  k_zero<<<(N_NODES + 255) / 256, 256, 0, stream>>>(cnt, N_NODES);
  k_deg <<<(N_EDGES + 255) / 256, 256, 0, stream>>>(dst, cnt);
  k_dinv<<<(N_NODES + 255) / 256, 256, 0, stream>>>(cnt, dinv);

  const float* Ws[3] = {W1, W2, W3};
  const float* bs[3] = {b1, b2, b3};
  const float* in = x;
  for (int l = 0; l < 3; ++l) {
    // H = act(in) @ W   (ReLU of previous aggregate fused into the tile load)
    k_gemm<<<N_NODES / 16, 256, 0, stream>>>(in, Ws[l], H, l > 0);
    // A = self-loop + bias, then scatter-add edge messages
    k_init<<<(N_NODES * D) / 256, 256, 0, stream>>>(H, dinv, bs[l], A);
    k_edge<<<(N_EDGES * 32) / 256, 256, 0, stream>>>(H, src, dst, dinv, A);
    in = A;  // next layer consumes pre-activation aggregate (ReLU fused in)
  }

  k_final<<<(N_TGT * 32) / 256, 256, 0, stream>>>(A, tgt, Wr, br, out);
}